// QeqSolver_69698729279846
// MI455X (gfx1250) — compile-verified
//
#include <hip/hip_runtime.h>
#include <hip/hip_bf16.h>
#include <math.h>

typedef __attribute__((ext_vector_type(2))) float v2f;
typedef __attribute__((ext_vector_type(8))) float v8f;

#define NATOM 256
#define SLDS 257          // LDS row stride: 257 mod 64 == 1 -> conflict-free column sweeps
#define NB 16             // 16 blocks of 16 along each dim
#define SQRT2_F  1.41421356f
#define SQRTPI_F 1.77245385f

// One workgroup (256 threads = 8 wave32) per charge-equilibration group.
// Whole 256x257 f32 matrix lives in LDS (264KB < 320KB/WGP).
__global__ __launch_bounds__(256)
void qeq_solve_kernel(const int*   __restrict__ node_attrs,
                      const float* __restrict__ chi,
                      const float* __restrict__ total_charge,
                      const float* __restrict__ positions,
                      const float* __restrict__ hardness,
                      const float* __restrict__ cov_radii,
                      float* __restrict__ out_q,
                      float* __restrict__ out_e)
{
    extern __shared__ float smem[];
    float* A   = smem;                 // NATOM * SLDS
    float* px  = A   + NATOM * SLDS;
    float* py  = px  + NATOM;
    float* pz  = py  + NATOM;
    float* s2  = pz  + NATOM;
    float* y1  = s2  + NATOM;          // RHS/solution 1: A x1 = -chi
    float* y2  = y1  + NATOM;          // RHS/solution 2: A x2 = 1
    float* red = y2  + NATOM;          // reduction scratch

    const int tid  = threadIdx.x;
    const int gid  = blockIdx.x;
    const int base = gid * NATOM;

    const int lane = tid & 31;
    const int wave = __builtin_amdgcn_readfirstlane(tid) >> 5;  // scalar wave id (8 waves)
    const int half = lane >> 4;    // 0 or 1
    const int l15  = lane & 15;

    // ---- per-atom data (thread t <-> atom t) ----
    const int   z    = node_attrs[base + tid];
    const float sig  = cov_radii[z];
    const float Ji   = hardness[z];
    const float chii = chi[base + tid];
    const float xi   = positions[(base + tid) * 3 + 0];
    const float yi   = positions[(base + tid) * 3 + 1];
    const float zi   = positions[(base + tid) * 3 + 2];
    px[tid] = xi; py[tid] = yi; pz[tid] = zi; s2[tid] = sig * sig;
    __syncthreads();

    // ---- Phase 1: build row `tid` of A in LDS ----
    {
        const float s2i = sig * sig;
        const int rowb = tid * SLDS;
        for (int j = 0; j < NATOM; ++j) {
            float v;
            if (j == tid) {
                v = Ji + 1.0f / (sig * SQRTPI_F);
            } else {
                const float dx = px[j] - xi, dy = py[j] - yi, dz = pz[j] - zi;
                const float r  = sqrtf(dx * dx + dy * dy + dz * dz);
                const float g  = sqrtf(s2i + s2[j]);
                v = erff(r / (SQRT2_F * g)) / r;
            }
            A[rowb + j] = v;
        }
    }

    // ---- Phase 2: blocked LU (no pivot), 16x16 tiles, WMMA trailing update ----
    for (int k = 0; k < NB; ++k) {
        const int k16 = k * 16;
        __syncthreads();

        // (a) factor diagonal block: wave 0 only, wave-synchronous (in-order LDS per wave)
        if (wave == 0) {
            for (int t = 0; t < 16; ++t) {
                const float dinv = 1.0f / A[(k16 + t) * SLDS + k16 + t];
                if (half == 0 && l15 > t)
                    A[(k16 + l15) * SLDS + k16 + t] *= dinv;    // L column t
                __builtin_amdgcn_wave_barrier();
                if (l15 > t) {
                    const float lrt = A[(k16 + l15) * SLDS + k16 + t];
                    for (int c = t + 1 + half; c < 16; c += 2)  // split columns across halves
                        A[(k16 + l15) * SLDS + k16 + c] -=
                            lrt * A[(k16 + t) * SLDS + k16 + c];
                }
                __builtin_amdgcn_wave_barrier();
            }
        }
        __syncthreads();

        const int nb = NB - 1 - k;
        if (nb > 0) {
            // (b) row panel U(k,j): forward subst vs unit-lower Lkk; one thread per column
            if (tid < nb * 16) {
                const int col = (k + 1 + (tid >> 4)) * 16 + (tid & 15);
                float u[16];
                #pragma unroll
                for (int t = 0; t < 16; ++t) {
                    float acc = A[(k16 + t) * SLDS + col];
                    for (int s = 0; s < t; ++s)
                        acc -= A[(k16 + t) * SLDS + (k16 + s)] * u[s];
                    u[t] = acc;
                    A[(k16 + t) * SLDS + col] = acc;
                }
            }
            // (c) column panel L(i,k): row solve vs non-unit upper Ukk; one thread per row
            if (tid < nb * 16) {
                const int row = (k + 1 + (tid >> 4)) * 16 + (tid & 15);
                float v[16];
                #pragma unroll
                for (int t = 0; t < 16; ++t) {
                    float acc = A[row * SLDS + k16 + t];
                    for (int s = 0; s < t; ++s)
                        acc -= v[s] * A[(k16 + s) * SLDS + (k16 + t)];
                    acc /= A[(k16 + t) * SLDS + (k16 + t)];
                    v[t] = acc;
                    A[row * SLDS + k16 + t] = acc;
                }
            }
            __syncthreads();

            // (d) trailing update: each wave owns a tile ROW (fixed bi) so the negated
            //     L fragments are loaded once and reused across all bj tiles.
            //     C(bi,bj) -= L(bi,k)*U(k,bj) via 4 chained V_WMMA_F32_16X16X4_F32.
            for (int bi_i = wave; bi_i < nb; bi_i += 8) {
                const int bi   = k + 1 + bi_i;
                const int arow = (bi * 16 + l15) * SLDS;   // A-frag: M=lane&15
                const int crow = bi * 16 + half * 8;       // C/D: VGPR v <-> M=v+8*half
                v2f afr[4];
                #pragma unroll
                for (int kk = 0; kk < 4; ++kk) {           // K pair = kk*4 + half*2
                    const int kq = k16 + kk * 4 + half * 2;
                    afr[kk].x = -A[arow + kq];
                    afr[kk].y = -A[arow + kq + 1];
                }
                for (int bj = k + 1; bj < NB; ++bj) {
                    const int ccol = bj * 16 + l15;        // N = lane&15
                    v8f cfr;
                    #pragma unroll
                    for (int v = 0; v < 8; ++v)
                        cfr[v] = A[(crow + v) * SLDS + ccol];
                    #pragma unroll
                    for (int kk = 0; kk < 4; ++kk) {
                        const int kq = k16 + kk * 4 + half * 2;
                        v2f bfr;
                        bfr.x = A[kq * SLDS + ccol];       // B: rows K=kq(+1), N=lane&15
                        bfr.y = A[(kq + 1) * SLDS + ccol];
                        cfr = __builtin_amdgcn_wmma_f32_16x16x4_f32(
                            false, afr[kk], false, bfr, (short)0, cfr, false, false);
                    }
                    #pragma unroll
                    for (int v = 0; v < 8; ++v)
                        A[(crow + v) * SLDS + ccol] = cfr[v];
                }
            }
        }
        __syncthreads();
    }

    // ---- Phase 3: blocked triangular solves, two RHS at once ----
    y1[tid] = -chii;
    y2[tid] = 1.0f;
    __syncthreads();

    // forward: unit-lower L
    for (int b = 0; b < NB; ++b) {
        const int j0 = b * 16;
        if (wave == 0) {                       // wave-synchronous diagonal-block solve
            for (int t = 0; t < 15; ++t) {
                const float v1 = y1[j0 + t];
                const float v2 = y2[j0 + t];
                if (half == 0 && l15 > t) {
                    const float l = A[(j0 + l15) * SLDS + j0 + t];
                    y1[j0 + l15] -= l * v1;
                    y2[j0 + l15] -= l * v2;
                }
                __builtin_amdgcn_wave_barrier();
            }
        }
        __syncthreads();
        if (tid >= j0 + 16) {                  // rank-16 update of the remaining rows
            float a1 = y1[tid], a2 = y2[tid];
            #pragma unroll
            for (int t = 0; t < 16; ++t) {
                const float l = A[tid * SLDS + j0 + t];
                a1 -= l * y1[j0 + t];
                a2 -= l * y2[j0 + t];
            }
            y1[tid] = a1; y2[tid] = a2;
        }
        __syncthreads();
    }

    // backward: upper U (non-unit diagonal)
    for (int b = NB - 1; b >= 0; --b) {
        const int j0 = b * 16;
        if (wave == 0) {                       // wave-synchronous diagonal-block solve
            for (int t = 15; t >= 0; --t) {
                const float dinv = 1.0f / A[(j0 + t) * SLDS + j0 + t];
                const float xv1  = y1[j0 + t] * dinv;
                const float xv2  = y2[j0 + t] * dinv;
                __builtin_amdgcn_wave_barrier();
                if (half == 0 && l15 < t) {
                    const float u = A[(j0 + l15) * SLDS + j0 + t];
                    y1[j0 + l15] -= u * xv1;
                    y2[j0 + l15] -= u * xv2;
                }
                if (half == 1 && l15 == t) { y1[j0 + t] = xv1; y2[j0 + t] = xv2; }
                __builtin_amdgcn_wave_barrier();
            }
        }
        __syncthreads();
        if (tid < j0) {                        // rank-16 update of rows above the block
            float a1 = y1[tid], a2 = y2[tid];
            #pragma unroll
            for (int t = 0; t < 16; ++t) {
                const float u = A[tid * SLDS + j0 + t];
                a1 -= u * y1[j0 + t];
                a2 -= u * y2[j0 + t];
            }
            y1[tid] = a1; y2[tid] = a2;
        }
        __syncthreads();
    }

    // ---- Phase 4: Lagrange multiplier, charges, energy ----
    auto block_reduce = [&](float v) -> float {
        red[tid] = v; __syncthreads();
        for (int s = 128; s > 0; s >>= 1) {
            if (tid < s) red[tid] += red[tid + s];
            __syncthreads();
        }
        const float r = red[0];
        __syncthreads();
        return r;
    };

    const float sum1 = block_reduce(y1[tid]);
    const float sum2 = block_reduce(y2[tid]);
    const float Qtot = total_charge[gid];
    const float lam  = (sum1 - Qtot) / sum2;
    const float q    = y1[tid] - lam * y2[tid];
    out_q[base + tid] = q;

    // E = 0.5 q^T A_off q + sum q^2/(2 sigma sqrtpi)
    //   = -0.5 q.chi - 0.5 lam Qtot - 0.5 sum J q^2   (using A q = -chi - lam*1)
    const float epart = -0.5f * q * chii - 0.5f * Ji * q * q;
    const float esum  = block_reduce(epart);
    if (tid == 0) out_e[gid] = esum - 0.5f * lam * Qtot;
}

extern "C" void kernel_launch(void* const* d_in, const int* in_sizes, int n_in,
                              void* d_out, int out_size, void* d_ws, size_t ws_size,
                              hipStream_t stream) {
    // setup_inputs order: 0=edge_index(unused) 1=node_attrs 2=chi 3=ptr(unused)
    //                     4=total_charge 5=positions 6=hardness 7=covalent_radii
    const int*   node_attrs = (const int*)  d_in[1];
    const float* chi        = (const float*)d_in[2];
    const float* tq         = (const float*)d_in[4];
    const float* pos        = (const float*)d_in[5];
    const float* hard       = (const float*)d_in[6];
    const float* cov        = (const float*)d_in[7];
    float* out = (float*)d_out;

    const int G    = in_sizes[4];  // 512 groups
    const int Ntot = in_sizes[2];  // 131072 atoms

    const size_t shmem = (size_t)(NATOM * SLDS + 7 * NATOM) * sizeof(float); // ~270 KB
    qeq_solve_kernel<<<G, 256, shmem, stream>>>(node_attrs, chi, tq, pos, hard, cov,
                                                out, out + Ntot);
}